// Kernel_15488212050779847048_53472342835829
// MI455X (gfx1250) — compile-verified
//
#include <hip/hip_runtime.h>

// CDNA5 / gfx1250, wave32. All matrix math via V_WMMA_F32_16X16X4_F32 (exact fp32).
// Register blocking: each wave computes a 2x2 grid of 16x16 tiles so every
// K=4 step issues 2 A + 2 B LDS b64 loads for 4 WMMAs (1:1 ds:wmma) and the
// 4 independent accumulator chains match the XDL pipeline depth.

typedef float v2f __attribute__((ext_vector_type(2)));
typedef float v8f __attribute__((ext_vector_type(8)));

#define NIMG 32
#define CDIM 128
#define K3C  384
#define HDIM 64
#define WDIM 64
#define HW   4096

#define ASTR 68   // LDS row stride (68 % 64 == 4 -> conflict-free b64 reads)

__device__ __forceinline__ v8f wmma4(v2f a, v2f b, v8f c) {
  // D = A(16x4,f32) * B(4x16,f32) + C(16x16,f32)
  return __builtin_amdgcn_wmma_f32_16x16x4_f32(
      /*neg_a=*/false, a, /*neg_b=*/false, b,
      /*c_mod=*/(short)0, c, /*reuse_a=*/false, /*reuse_b=*/false);
}

// ---------------------------------------------------------------------------
// Stages 1, 2, 4: per image, D[128 x 4096] = A[128 x 384] * B[384 x 4096],
// with B synthesized on the fly (never materialized):
//   MODE 1: A = w3,  B[kk=3c+k][p] = t1  (x tap at w+2k-2)            -> t3
//   MODE 2: A = w6,  B[kk=3i+r][p] = t5 shifted by 3r-3 rows          -> t6
//   MODE 4: A = t9n, B[kk=3c+k][p] = t12 from x,t3,t6,p12             -> out
// Block: 256 thr (8 waves); covers full M=128 x 64 columns.
// Wave w: M-tiles 2*(w&3) (rows 32(w&3)..+32), N-tiles at cols 32*(w>>2)..+32.
// K staged in 6 chunks of 64 through LDS.
// ---------------------------------------------------------------------------
template <int MODE>
__global__ __launch_bounds__(256) void gemm_wide(
    const float* __restrict__ x, const float* __restrict__ Amat,
    const float* __restrict__ p5, const float* __restrict__ t3,
    const float* __restrict__ t6, const float* __restrict__ p12,
    float* __restrict__ D, float scale)
{
  __shared__ __align__(16) float As[128 * ASTR];  // [m][k-chunk]
  __shared__ __align__(16) float Bs[64 * ASTR];   // [col][k-chunk]

  const int n    = blockIdx.x >> 6;          // image
  const int p0   = (blockIdx.x & 63) << 6;   // first of 64 columns
  const int tid  = threadIdx.x;
  const int lane = tid & 31;
  const int wave = tid >> 5;
  const int mt   = (wave & 3) << 5;          // first of 32 rows for this wave
  const int nt   = (wave >> 2) << 4;         // col tile pair: 0 or 32 -> <<5?  see below
  const int ntc  = (wave >> 2) << 5;         // first of 32 cols for this wave
  const int lhi  = lane >> 4, llo = lane & 15;
  (void)nt;

  const float* Ab = (MODE == 4) ? (Amat + (size_t)n * CDIM * K3C) : Amat;

  v8f acc00 = {0.f, 0.f, 0.f, 0.f, 0.f, 0.f, 0.f, 0.f};
  v8f acc01 = acc00, acc10 = acc00, acc11 = acc00;

  for (int kc = 0; kc < 6; ++kc) {
    const int kk0 = kc << 6;

    // --- stage A chunk [128 x 64] into LDS ---
    for (int i = tid; i < 128 * 64; i += 256) {
      int m = i >> 6, k = i & 63;
      As[m * ASTR + k] = Ab[m * K3C + kk0 + k];
    }

    // --- synthesize B chunk [64 cols x 64 K] into LDS ---
    for (int i = tid; i < 64 * 64; i += 256) {
      int col = i & 63, kl = i >> 6;
      int kk = kk0 + kl;
      int c = kk / 3, k = kk - 3 * c;           // channel, tap
      int p = p0 + col;
      int h = p >> 6, w = p & 63;
      float val;
      if (MODE == 1) {                          // t1: w-tap of x, zero padded
        int w2 = w + 2 * k - 2;
        val = (w2 >= 0 && w2 < WDIM)
                  ? x[(size_t)(n * CDIM + c) * HW + (h << 6) + w2] : 0.f;
      } else if (MODE == 2) {                   // t5 shifted by 3k-3 rows
        int h2 = h + 3 * k - 3;
        if (h2 >= 0 && h2 < HDIM) {
          const float* xr = x + (size_t)(n * CDIM + c) * HW + (h2 << 6);
          float m3 = xr[w];                     // t2 = max of taps (pad = 0)
          m3 = fmaxf(m3, (w >= 2) ? xr[w - 2] : 0.f);
          m3 = fmaxf(m3, (w < WDIM - 2) ? xr[w + 2] : 0.f);
          val = p5[(h2 << 6) + w] * m3;         // t5
        } else {
          val = 0.f;
        }
      } else {                                  // MODE 4: t12
        size_t idx = (size_t)(n * CDIM + c) * HW + p;
        float xv = x[idx], t6v = t6[idx], t3v = t3[idx];
        int w2 = w + 2 * k - 2;
        float t1v = (w2 >= 0 && w2 < WDIM) ? x[idx - w + w2] : 0.f;
        float t7  = fmaxf(xv, t6v);
        float t11 = fmaxf(t6v, t7 + fmaxf(t3v, t1v));  // max(t6, t7 + t4)
        val = p12[k * HDIM + h] * t11;
      }
      Bs[col * ASTR + kl] = val;
    }
    __syncthreads();

    // --- 16 WMMA K-steps of 4: 2 A + 2 B loads feed 4 WMMAs each ---
    const int a0 = (mt + llo) * ASTR + 2 * lhi;
    const int a1 = (mt + 16 + llo) * ASTR + 2 * lhi;
    const int b0 = (ntc + llo) * ASTR + 2 * lhi;
    const int b1 = (ntc + 16 + llo) * ASTR + 2 * lhi;
#pragma unroll
    for (int s = 0; s < 16; ++s) {
      v2f va0 = *(const v2f*)&As[a0 + 4 * s];
      v2f va1 = *(const v2f*)&As[a1 + 4 * s];
      v2f vb0 = *(const v2f*)&Bs[b0 + 4 * s];
      v2f vb1 = *(const v2f*)&Bs[b1 + 4 * s];
      acc00 = wmma4(va0, vb0, acc00);
      acc01 = wmma4(va0, vb1, acc01);
      acc10 = wmma4(va1, vb0, acc10);
      acc11 = wmma4(va1, vb1, acc11);
    }
    __syncthreads();
  }

  // --- store the 2x2 tile grid: rows = channels, cols = contiguous pixels ---
#pragma unroll
  for (int r = 0; r < 8; ++r) {
    int m0 = mt + r + 8 * lhi;
    float* row0 = D + (size_t)(n * CDIM + m0) * HW + p0 + ntc;
    float* row1 = D + (size_t)(n * CDIM + m0 + 16) * HW + p0 + ntc;
    row0[llo]      = acc00[r] * scale;
    row0[llo + 16] = acc01[r] * scale;
    row1[llo]      = acc10[r] * scale;
    row1[llo + 16] = acc11[r] * scale;
  }
}

// ---------------------------------------------------------------------------
// Stage 3: t9[n][c][kk] = (1/64) * sum_p t8[c,p] * t1[kk,p]
// GEMM per image: M=128 (c), N=384 (kk), K=4096 (p).
// Block: full M x 64 kk-columns; K staged in 64-wide p-chunks through LDS.
// A = t8 = p8[c]*max(x,t6) synthesized; B = t1 synthesized (both fused).
// Same 2x2 register blocking as gemm_wide.
// ---------------------------------------------------------------------------
__global__ __launch_bounds__(256) void gemm_t9(
    const float* __restrict__ x, const float* __restrict__ t6,
    const float* __restrict__ p8, float* __restrict__ t9)
{
  __shared__ __align__(16) float As[128 * ASTR];  // [c][p-chunk]
  __shared__ __align__(16) float Bs[64 * ASTR];   // [kk-col][p-chunk]

  const int n    = blockIdx.x / 6;
  const int kk0  = (blockIdx.x - n * 6) << 6;     // 6 chunks of 64 kk
  const int tid  = threadIdx.x;
  const int lane = tid & 31;
  const int wave = tid >> 5;
  const int mt   = (wave & 3) << 5;
  const int ntc  = (wave >> 2) << 5;
  const int lhi  = lane >> 4, llo = lane & 15;

  v8f acc00 = {0.f, 0.f, 0.f, 0.f, 0.f, 0.f, 0.f, 0.f};
  v8f acc01 = acc00, acc10 = acc00, acc11 = acc00;

  for (int pc = 0; pc < HW; pc += 64) {
    // A chunk: t8[c][pc..pc+64)
    for (int i = tid; i < 128 * 64; i += 256) {
      int c = i >> 6, kp = i & 63;
      size_t idx = (size_t)(n * CDIM + c) * HW + pc + kp;
      As[c * ASTR + kp] = p8[c] * fmaxf(x[idx], t6[idx]);
    }
    // B chunk: t1[kk0+col][pc..pc+64)
    for (int i = tid; i < 64 * 64; i += 256) {
      int col = i & 63, kp = i >> 6;
      int kk = kk0 + col;
      int c = kk / 3, k = kk - 3 * c;
      int p = pc + kp;
      int w = p & 63;
      int w2 = w + 2 * k - 2;
      Bs[col * ASTR + kp] =
          (w2 >= 0 && w2 < WDIM) ? x[(size_t)(n * CDIM + c) * HW + p - w + w2]
                                 : 0.f;
    }
    __syncthreads();

    const int a0 = (mt + llo) * ASTR + 2 * lhi;
    const int a1 = (mt + 16 + llo) * ASTR + 2 * lhi;
    const int b0 = (ntc + llo) * ASTR + 2 * lhi;
    const int b1 = (ntc + 16 + llo) * ASTR + 2 * lhi;
#pragma unroll
    for (int s = 0; s < 16; ++s) {
      v2f va0 = *(const v2f*)&As[a0 + 4 * s];
      v2f va1 = *(const v2f*)&As[a1 + 4 * s];
      v2f vb0 = *(const v2f*)&Bs[b0 + 4 * s];
      v2f vb1 = *(const v2f*)&Bs[b1 + 4 * s];
      acc00 = wmma4(va0, vb0, acc00);
      acc01 = wmma4(va0, vb1, acc01);
      acc10 = wmma4(va1, vb0, acc10);
      acc11 = wmma4(va1, vb1, acc11);
    }
    __syncthreads();
  }

#pragma unroll
  for (int r = 0; r < 8; ++r) {
    int m0 = mt + r + 8 * lhi;
    float* row0 = t9 + (size_t)(n * CDIM + m0) * K3C + kk0 + ntc;
    float* row1 = t9 + (size_t)(n * CDIM + m0 + 16) * K3C + kk0 + ntc;
    row0[llo]      = acc00[r] * 0.015625f;   // 1/sqrt(HW) = 1/64
    row0[llo + 16] = acc01[r] * 0.015625f;
    row1[llo]      = acc10[r] * 0.015625f;
    row1[llo + 16] = acc11[r] * 0.015625f;
  }
}

extern "C" void kernel_launch(void* const* d_in, const int* in_sizes, int n_in,
                              void* d_out, int out_size, void* d_ws, size_t ws_size,
                              hipStream_t stream)
{
  const float* x   = (const float*)d_in[0];
  const float* w3  = (const float*)d_in[1];
  const float* p5  = (const float*)d_in[2];
  const float* w6  = (const float*)d_in[3];
  const float* p8  = (const float*)d_in[4];
  const float* p12 = (const float*)d_in[5];
  float* out = (float*)d_out;

  float* t3 = (float*)d_ws;                          //  67 MB
  float* t6 = t3 + (size_t)NIMG * CDIM * HW;         //  67 MB
  float* t9 = t6 + (size_t)NIMG * CDIM * HW;         // 6.3 MB

  const float inv_sqrt_3c = 0.051031036307982884f;   // 1/sqrt(384)

  dim3 blk(256);
  // Stage 1: t3 = w3 . T1
  gemm_wide<1><<<dim3(2048), blk, 0, stream>>>(x, w3, nullptr, nullptr, nullptr,
                                               nullptr, t3, 1.0f);
  // Stage 2: t6 = w6 . T5taps
  gemm_wide<2><<<dim3(2048), blk, 0, stream>>>(x, w6, p5, nullptr, nullptr,
                                               nullptr, t6, 1.0f);
  // Stage 3: t9 = t8 . T1^T / 64
  gemm_t9<<<dim3(192), blk, 0, stream>>>(x, t6, p8, t9);
  // Stage 4: out[c][p] = (t9 . T12) / sqrt(384)
  gemm_wide<4><<<dim3(2048), blk, 0, stream>>>(x, t9, nullptr, t3, t6, p12, out,
                                               inv_sqrt_3c);
}